// SwitchMOE_67010079752532
// MI455X (gfx1250) — compile-verified
//
#include <hip/hip_runtime.h>
#include <hip/hip_bf16.h>
#include <math.h>

// Problem constants (match reference)
#define BB 8
#define TT 2048
#define DD 1024
#define FF 4096
#define EE 16
#define TOK (BB*TT)          // 16384
#define CAP 1280             // int(1.25 * 16384 / 16)
#define KSLOT 16             // K = 13 kept slots, padded to 16 rows

typedef float v2f __attribute__((ext_vector_type(2)));
typedef float v4f __attribute__((ext_vector_type(4)));
typedef float v8f __attribute__((ext_vector_type(8)));

// Two WMMA K-steps fed by one float4 per lane (see analysis: K-permutation
// is legal because GEMM K-accumulation is commutative and A/B share the
// same lane->memory-column mapping).
#define WMMA_PAIR(acc, a4, b4)                                            \
    do {                                                                  \
        v2f _a0 = { (a4).x, (a4).y }, _b0 = { (b4).x, (b4).y };           \
        v2f _a1 = { (a4).z, (a4).w }, _b1 = { (b4).z, (b4).w };           \
        (acc) = __builtin_amdgcn_wmma_f32_16x16x4_f32(                    \
            false, _a0, false, _b0, (short)0, (acc), false, false);       \
        (acc) = __builtin_amdgcn_wmma_f32_16x16x4_f32(                    \
            false, _a1, false, _b1, (short)0, (acc), false, false);       \
    } while (0)

// ---------------------------------------------------------------------------
// 0) zero the output (scatter target) — poisoned by harness, must re-zero
// ---------------------------------------------------------------------------
__global__ void zero_k(float* __restrict__ p, size_t n) {
    size_t i = (size_t)blockIdx.x * blockDim.x + threadIdx.x;
    size_t stride = (size_t)gridDim.x * blockDim.x;
    for (; i < n; i += stride) p[i] = 0.0f;
}

// ---------------------------------------------------------------------------
// 1) Router: logits[t,e] = x[t,:] . Wr[e,:]  via V_WMMA_F32_16X16X4_F32
//    One wave per 16-token tile; N=16 == E exactly.
// ---------------------------------------------------------------------------
__global__ __launch_bounds__(32) void router_k(const float* __restrict__ x,
                                               const float* __restrict__ Wr,
                                               float* __restrict__ logits) {
    const int t0   = blockIdx.x * 16;
    const int lane = threadIdx.x;
    const int grp  = lane >> 4;       // 0 or 1
    const int nm   = lane & 15;       // A: row M; B: col N (expert)

    const float* __restrict__ ar = x  + (size_t)(t0 + nm) * DD + grp * 4;
    const float* __restrict__ br = Wr + (size_t)nm * DD        + grp * 4;

    v8f acc = {0.f,0.f,0.f,0.f,0.f,0.f,0.f,0.f};
    for (int d = 0; d < DD; d += 8) {
        v4f a = __builtin_nontemporal_load((const v4f*)(ar + d));  // x: single use
        v4f b = *(const v4f*)(br + d);                             // Wr: tiny, cached
        WMMA_PAIR(acc, a, b);
    }
    // C/D layout: lane<16 -> M=i, lane>=16 -> M=8+i ; N = lane&15
    for (int i = 0; i < 8; ++i) {
        int m = grp * 8 + i;
        logits[(size_t)(t0 + m) * EE + nm] = acc[i];
    }
}

// ---------------------------------------------------------------------------
// 2) Softmax / top1 / top1p + deterministic per-block expert prob sums
// ---------------------------------------------------------------------------
__global__ __launch_bounds__(256) void softmax_k(const float* __restrict__ logits,
                                                 int* __restrict__ top1,
                                                 float* __restrict__ top1p,
                                                 float* __restrict__ bsum) {
    const int tok = blockIdx.x * 256 + threadIdx.x;
    const float* lr = logits + (size_t)tok * EE;
    float l[EE];
    for (int e = 0; e < EE; ++e) l[e] = lr[e];

    float mx = l[0]; int bi = 0;
    for (int e = 1; e < EE; ++e) if (l[e] > mx) { mx = l[e]; bi = e; }  // first max

    float p[EE]; float s = 0.f;
    for (int e = 0; e < EE; ++e) { p[e] = expf(l[e] - mx); s += p[e]; }
    const float inv = 1.0f / s;
    for (int e = 0; e < EE; ++e) p[e] *= inv;

    top1[tok]  = bi;
    top1p[tok] = p[bi];

    // fixed-order tree reduction per expert -> bsum[block][e] (no atomics)
    __shared__ float red[256];
    for (int e = 0; e < EE; ++e) {
        red[threadIdx.x] = p[e];
        __syncthreads();
        for (int s2 = 128; s2 > 0; s2 >>= 1) {
            if (threadIdx.x < s2) red[threadIdx.x] += red[threadIdx.x + s2];
            __syncthreads();
        }
        if (threadIdx.x == 0) bsum[blockIdx.x * EE + e] = red[0];
        __syncthreads();
    }
}

// ---------------------------------------------------------------------------
// 3) Ordered rank scan: keep tokens whose per-expert rank % CAP == 0.
//    Single block: chunk histograms -> exclusive prefix -> rescan.
//    Also computes lb_loss deterministically.
// ---------------------------------------------------------------------------
__global__ __launch_bounds__(256) void rank_k(const int* __restrict__ top1,
                                              const float* __restrict__ bsum,
                                              int* __restrict__ idxArr,
                                              int* __restrict__ validArr,
                                              float* __restrict__ loss_out) {
    __shared__ int hist[256][EE];
    __shared__ int offs[256][EE];
    __shared__ int totals[EE];
    const int t = threadIdx.x;

    idxArr[t] = 0;            // 256 slots (E*KSLOT)
    validArr[t] = 0;

    int cnt[EE];
    for (int e = 0; e < EE; ++e) cnt[e] = 0;
    for (int i = 0; i < 64; ++i) cnt[top1[t * 64 + i]]++;
    for (int e = 0; e < EE; ++e) hist[t][e] = cnt[e];
    __syncthreads();

    if (t < EE) {
        int run = 0;
        for (int c = 0; c < 256; ++c) { offs[c][t] = run; run += hist[c][t]; }
        totals[t] = run;
    }
    __syncthreads();

    for (int e = 0; e < EE; ++e) cnt[e] = offs[t][e];
    for (int i = 0; i < 64; ++i) {
        int tok = t * 64 + i;
        int e = top1[tok];
        int r = cnt[e];
        if ((r % CAP) == 0) {
            int slot = r / CAP;
            if (slot < KSLOT) { idxArr[e * KSLOT + slot] = tok;
                                validArr[e * KSLOT + slot] = 1; }
        }
        cnt[e] = r + 1;
    }

    if (t == 0) {
        float lb = 0.f;
        for (int e = 0; e < EE; ++e) {
            float ps = 0.f;
            for (int b = 0; b < 64; ++b) ps += bsum[b * EE + e];
            lb += ((float)totals[e] / (float)TOK) * (ps / (float)TOK);
        }
        *loss_out = lb * (float)EE * 0.01f;
    }
}

// ---------------------------------------------------------------------------
// 4) FFN layer 1: h[e,m,f] = gelu( xg[e,m,:] . W1[e,f,:] + b1[e,f] )
//    4 waves/WG, each wave one 16x16 tile; invalid rows zeroed via A mask.
//    W1 streamed non-temporally (single use, 268 MB).
// ---------------------------------------------------------------------------
__global__ __launch_bounds__(128) void ffn1_k(const float* __restrict__ x,
                                              const float* __restrict__ W1,
                                              const float* __restrict__ b1,
                                              const int* __restrict__ idxArr,
                                              const int* __restrict__ validArr,
                                              float* __restrict__ hbuf) {
    const int e    = blockIdx.y;
    const int wave = threadIdx.x >> 5;
    const int lane = threadIdx.x & 31;
    const int grp  = lane >> 4;
    const int nm   = lane & 15;
    const int fb   = blockIdx.x * 64 + wave * 16;

    const int   tok = idxArr[e * KSLOT + nm];
    const float am  = validArr[e * KSLOT + nm] ? 1.0f : 0.0f;  // zero-pad rows
    const float* __restrict__ ar = x  + (size_t)tok * DD + grp * 4;
    const float* __restrict__ br = W1 + ((size_t)e * FF + (size_t)(fb + nm)) * DD + grp * 4;

    v8f acc = {0.f,0.f,0.f,0.f,0.f,0.f,0.f,0.f};
    for (int d = 0; d < DD; d += 8) {
        v4f a = *(const v4f*)(ar + d);              // gathered rows: L2-resident
        a.x *= am; a.y *= am; a.z *= am; a.w *= am; // mask keeps EXEC uniform
        v4f b = __builtin_nontemporal_load((const v4f*)(br + d));
        WMMA_PAIR(acc, a, b);
    }

    const int   f    = fb + nm;
    const float bias = b1[e * FF + f];
    for (int i = 0; i < 8; ++i) {
        int m = grp * 8 + i;
        float v = acc[i] + bias;
        float g = 0.5f * v * (1.0f + erff(v * 0.70710678118654752f)); // exact gelu
        hbuf[((size_t)e * 16 + m) * FF + f] = g;
    }
}

// ---------------------------------------------------------------------------
// 5) FFN layer 2 + scale + scatter:
//    out[tok, dcol] = (h[e,m,:] . W2[e,dcol,:] + b2[e,dcol]) * top1p[tok]
//    Unique writer per token (top-1 routing, unique ranks) -> plain stores.
//    W2 streamed non-temporally (single use, 268 MB); hbuf reused from L2.
// ---------------------------------------------------------------------------
__global__ __launch_bounds__(128) void ffn2_k(const float* __restrict__ hbuf,
                                              const float* __restrict__ W2,
                                              const float* __restrict__ b2,
                                              const int* __restrict__ idxArr,
                                              const int* __restrict__ validArr,
                                              const float* __restrict__ top1p,
                                              float* __restrict__ out) {
    const int e    = blockIdx.y;
    const int wave = threadIdx.x >> 5;
    const int lane = threadIdx.x & 31;
    const int grp  = lane >> 4;
    const int nm   = lane & 15;
    const int db   = blockIdx.x * 64 + wave * 16;

    const float* __restrict__ ar = hbuf + ((size_t)e * 16 + nm) * FF + grp * 4;
    const float* __restrict__ br = W2   + ((size_t)e * DD + (size_t)(db + nm)) * FF + grp * 4;

    v8f acc = {0.f,0.f,0.f,0.f,0.f,0.f,0.f,0.f};
    for (int f = 0; f < FF; f += 8) {
        v4f a = *(const v4f*)(ar + f);              // hbuf: 4 MB, L2-resident
        v4f b = __builtin_nontemporal_load((const v4f*)(br + f));
        WMMA_PAIR(acc, a, b);
    }

    const int   dcol = db + nm;
    const float bias = b2[e * DD + dcol];
    for (int i = 0; i < 8; ++i) {
        int m = grp * 8 + i;
        if (validArr[e * KSLOT + m]) {
            int tok = idxArr[e * KSLOT + m];
            out[(size_t)tok * DD + dcol] = (acc[i] + bias) * top1p[tok];
        }
    }
}

// ---------------------------------------------------------------------------
extern "C" void kernel_launch(void* const* d_in, const int* in_sizes, int n_in,
                              void* d_out, int out_size, void* d_ws, size_t ws_size,
                              hipStream_t stream) {
    const float* x  = (const float*)d_in[0];   // [TOK, D]
    const float* Wr = (const float*)d_in[1];   // [E, D]
    const float* W1 = (const float*)d_in[2];   // [E, F, D]
    const float* b1 = (const float*)d_in[3];   // [E, F]
    const float* W2 = (const float*)d_in[4];   // [E, D, F]
    const float* b2 = (const float*)d_in[5];   // [E, D]
    float* out = (float*)d_out;                // [TOK*D] out + [1] lb_loss

    // workspace layout (floats)
    float* ws      = (float*)d_ws;
    float* logits  = ws;                               // 262144
    int*   top1    = (int*)  (ws + 262144);            // 16384
    float* top1p   =          ws + 262144 + 16384;     // 16384
    float* bsum    =          ws + 262144 + 32768;     // 64*16 = 1024
    int*   idxA    = (int*)  (ws + 295936);            // 256
    int*   validA  = (int*)  (ws + 296192);            // 256
    float* hbuf    =          ws + 296448;             // E*16*F = 1048576

    zero_k<<<4096, 256, 0, stream>>>(out, (size_t)out_size);

    router_k<<<TOK / 16, 32, 0, stream>>>(x, Wr, logits);

    softmax_k<<<TOK / 256, 256, 0, stream>>>(logits, top1, top1p, bsum);

    rank_k<<<1, 256, 0, stream>>>(top1, bsum, idxA, validA, out + (out_size - 1));

    ffn1_k<<<dim3(FF / 64, EE), 128, 0, stream>>>(x, W1, b1, idxA, validA, hbuf);

    ffn2_k<<<dim3(DD / 64, EE), 128, 0, stream>>>(hbuf, W2, b2, idxA, validA, top1p, out);
}